// Nms3d_63737314673336
// MI455X (gfx1250) — compile-verified
//
#include <hip/hip_runtime.h>
#include <stdint.h>

#define N_BOX 4096
#define IOU_TR 0.3f
#define EPS_F  1e-7f
// iou > t  <=>  inter > t*(vs - inter + eps)  <=>  inter > (t/(1+t))*vs + (t/(1+t))*eps
#define C_SCALE (IOU_TR / (1.0f + IOU_TR))   // 3/13
#define C_EPS   (C_SCALE * EPS_F)

typedef __attribute__((ext_vector_type(2))) float v2f;
typedef __attribute__((ext_vector_type(8))) float v8f;

// -------------------------------------------------------------------------
// Kernel 1: stable descending sort by score (bitonic, in-LDS), then gather
// sorted rows. key = (score_bits << 32) | (0xFFFFFFFF - idx): scores are
// non-negative so the raw f32 bit pattern is order-monotone; low bits give
// stability (smaller original index wins ties when sorting descending).
// -------------------------------------------------------------------------
__global__ __launch_bounds__(1024) void nms_sort_kernel(
    const float* __restrict__ results, float* __restrict__ sortedRows) {
  __shared__ unsigned long long keys[N_BOX];
  const int tid = threadIdx.x;
  for (int i = tid; i < N_BOX; i += 1024) {
    unsigned sb = __float_as_uint(results[(size_t)i * 8]);
    keys[i] = ((unsigned long long)sb << 32) |
              (unsigned long long)(0xFFFFFFFFu - (unsigned)i);
  }
  __syncthreads();
  for (int k = 2; k <= N_BOX; k <<= 1) {
    for (int j = k >> 1; j > 0; j >>= 1) {
      for (int i = tid; i < N_BOX; i += 1024) {
        int ixj = i ^ j;
        if (ixj > i) {
          unsigned long long a = keys[i], b = keys[ixj];
          bool desc = ((i & k) == 0);
          bool sw = desc ? (a < b) : (a > b);
          if (sw) { keys[i] = b; keys[ixj] = a; }
        }
      }
      __syncthreads();
    }
  }
  for (int p = tid; p < N_BOX; p += 1024) {
    unsigned o = 0xFFFFFFFFu - (unsigned)(keys[p] & 0xFFFFFFFFull);
    const float4* src = (const float4*)(results + (size_t)o * 8);
    float4* dst = (float4*)(sortedRows + (size_t)p * 8);
    dst[0] = src[0];
    dst[1] = src[1];
  }
}

// -------------------------------------------------------------------------
// Kernel 2: suppression bitmask, one wave32 per 16x16 tile.
// mask16[row][colTile] bit c  <=>  iou(row, col) > 0.3 AND col > row.
// - Lower-triangle tiles store zeros and exit (ws is poisoned, so the
//   stores are mandatory, but half the grid skips all loads + math).
// - One v_wmma_f32_16x16x4_f32 computes (3/13)*(vol_i + vol_j) for all 256
//   pairs, i.e. the *pre-scaled threshold base*: the scale is folded into
//   the A operand, so the per-pair tail is just add-const + compare.
// - col>row only matters on exact-diagonal tiles (bases are 16-aligned and
//   colBase < rowBase was excluded), handled by constant ballot masks
//   under a uniform branch instead of a per-pair integer compare.
// -------------------------------------------------------------------------
#if __has_builtin(__builtin_amdgcn_wmma_f32_16x16x4_f32)
#define NMS_HAVE_WMMA4 1
#endif

__global__ __launch_bounds__(32) void nms_mask_kernel(
    const float* __restrict__ sorted, unsigned short* __restrict__ mask16) {
  const int lane = threadIdx.x;
  const int rowBase = blockIdx.y * 16;
  const int colBase = blockIdx.x * 16;

  if (colBase < rowBase) {  // all-zero tile: store & leave
    if ((lane & 15) == 0) {
      const int r = rowBase + ((lane >> 4) << 3);
#pragma unroll
      for (int v = 0; v < 8; ++v)
        mask16[(size_t)(r + v) * 256 + blockIdx.x] = 0;
    }
    return;
  }

  __shared__ float rlo[16][3];
  __shared__ float rhi[16][3];
  __shared__ float rvolScaled[16];  // (3/13) * vol_row

  if (lane < 16) {
    const float* rp = sorted + (size_t)(rowBase + lane) * 8;
    float l0 = rp[2], l1 = rp[3], l2 = rp[4];
    float h0 = rp[5], h1 = rp[6], h2 = rp[7];
    rlo[lane][0] = l0; rlo[lane][1] = l1; rlo[lane][2] = l2;
    rhi[lane][0] = h0; rhi[lane][1] = h1; rhi[lane][2] = h2;
    rvolScaled[lane] = C_SCALE * ((h0 - l0) * (h1 - l1) * (h2 - l2));
  }
  const int n = lane & 15;
  const float* cp = sorted + (size_t)(colBase + n) * 8;
  float cl0 = cp[2], cl1 = cp[3], cl2 = cp[4];
  float ch0 = cp[5], ch1 = cp[6], ch2 = cp[7];
  float cvol = (ch0 - cl0) * (ch1 - cl1) * (ch2 - cl2);
  __syncthreads();

#ifdef NMS_HAVE_WMMA4
  // A (16x4): row m = [c*vol_m, c, 0, 0]; B (4x16): col n = [1, vol_n, 0, 0]^T
  // D[m][n] = c*vol_m + c*vol_n  (c = 3/13), exactly in the (lane=col,
  // vgpr=row) layout consumed below.
  v2f a, b;
  a.x = (lane < 16) ? rvolScaled[lane & 15] : 0.0f;
  a.y = (lane < 16) ? C_SCALE : 0.0f;
  b.x = (lane < 16) ? 1.0f : 0.0f;
  b.y = (lane < 16) ? cvol : 0.0f;
  v8f cacc = {};
  v8f thrBase = __builtin_amdgcn_wmma_f32_16x16x4_f32(
      false, a, false, b, (short)0, cacc, false, false);
#else
  float ccvol = C_SCALE * cvol;
#endif

  const bool diag = (colBase == rowBase);
  unsigned bal[8];

#pragma unroll
  for (int v = 0; v < 8; ++v) {
    const int m = v + ((lane >> 4) << 3);  // C-layout row for this lane/vgpr
    float rl0 = rlo[m][0], rl1 = rlo[m][1], rl2 = rlo[m][2];
    float rh0 = rhi[m][0], rh1 = rhi[m][1], rh2 = rhi[m][2];
#ifdef NMS_HAVE_WMMA4
    float rhs = thrBase[v] + C_EPS;        // c*(vol_m+vol_n) + c*eps
#else
    float rhs = (rvolScaled[m] + ccvol) + C_EPS;
#endif
    float ix = fminf(rh0, ch0) - fmaxf(rl0, cl0);
    float iy = fminf(rh1, ch1) - fmaxf(rl1, cl1);
    float iz = fminf(rh2, ch2) - fmaxf(rl2, cl2);
    float inter = fmaxf(ix, 0.0f) * fmaxf(iy, 0.0f) * fmaxf(iz, 0.0f);
    unsigned w = (unsigned)__ballot(inter > rhs);  // wave32: bit l = lane l
    if (diag) {
      // keep only col > row: low half row = v, high half row = v+8
      const unsigned mlo = (0xFFFFu << (v + 1)) & 0xFFFFu;
      const unsigned mhi = (0xFFFFu << (v + 9)) & 0xFFFFu;
      w &= mlo | (mhi << 16);
    }
    bal[v] = w;
  }

  // Single divergence region for all 16 halfword stores.
  if ((lane & 15) == 0) {
    const int half = lane >> 4;  // 0: bits[15:0] rows v; 1: bits[31:16] rows v+8
    const int r = rowBase + half * 8;
#pragma unroll
    for (int v = 0; v < 8; ++v) {
      unsigned hw = half ? (bal[v] >> 16) : (bal[v] & 0xFFFFu);
      mask16[(size_t)(r + v) * 256 + blockIdx.x] = (unsigned short)hw;
    }
  }
}

// -------------------------------------------------------------------------
// Kernel 3: greedy scan, single wave32. The 4096-bit "suppressed" vector
// lives in registers (word w owned by lane w&31, slot w>>5). Rows are
// consumed in 32-row chunks, double-buffered into LDS with CDNA5 async
// global->LDS loads (ASYNCcnt) so the serial chain only touches LDS/regs.
// -------------------------------------------------------------------------
__device__ __forceinline__ unsigned lds_off_u32(const void* p) {
  // Generic LDS pointer: addr[31:0] is the byte offset from the wave's
  // LDS base (ISA §10.2 aperture mapping) — exactly what the async
  // instruction's LDS-address VGPR expects.
  return (unsigned)(unsigned long long)(uintptr_t)p;
}

__global__ __launch_bounds__(32) void nms_scan_kernel(
    const unsigned* __restrict__ mask32, unsigned* __restrict__ keptOut) {
  __shared__ unsigned buf[2][32 * 128];  // 2 x 16KB chunk buffers
  const int lane = threadIdx.x;
  unsigned rem0 = 0, rem1 = 0, rem2 = 0, rem3 = 0;

  auto issue_chunk = [&](int c, int s) {
    const uint4* g = (const uint4*)(mask32 + (size_t)c * 4096);
    unsigned base = lds_off_u32(&buf[s][0]);
#pragma unroll 8
    for (int t = 0; t < 32; ++t) {  // 32 x (32 lanes x 16B) = 16KB
      unsigned idx4 = (unsigned)(t * 32 + lane);
      const uint4* gp = g + idx4;
      unsigned loff = base + idx4 * 16u;
      asm volatile("global_load_async_to_lds_b128 %0, %1, off"
                   :: "v"(loff), "v"(gp) : "memory");
    }
  };

  issue_chunk(0, 0);
  for (int b = 0; b < 128; ++b) {
    const int s = b & 1;
    if (b + 1 < 128) {
      issue_chunk(b + 1, s ^ 1);
      // 64 outstanding max; async loads complete in order, so <=32 left
      // means all 32 of chunk b have landed in LDS.
      asm volatile("s_wait_asynccnt 0x20" ::: "memory");
    } else {
      asm volatile("s_wait_asynccnt 0x0" ::: "memory");
    }
    const unsigned* bufc = &buf[s][0];
    // word b (this chunk's own columns) of row (32b + lane)
    unsigned myrow = bufc[lane * 128 + b];

    unsigned ownerVal;
    switch (b >> 5) {
      case 0: ownerVal = rem0; break;
      case 1: ownerVal = rem1; break;
      case 2: ownerVal = rem2; break;
      default: ownerVal = rem3; break;
    }
    unsigned cur = (unsigned)__shfl((int)ownerVal, b & 31);

    // Serial 32-step recurrence, shuffles only (no memory, no barriers).
    unsigned keptw = 0;
    for (int i = 0; i < 32; ++i) {
      unsigned m = (unsigned)__shfl((int)myrow, i);
      if (((cur >> i) & 1u) == 0u) {  // row 32b+i still alive -> keep it
        keptw |= (1u << i);
        cur |= m;                     // suppress later rows in this chunk
      }
    }
    if (lane == (b & 31)) {
      switch (b >> 5) {
        case 0: rem0 = cur; break;
        case 1: rem1 = cur; break;
        case 2: rem2 = cur; break;
        default: rem3 = cur; break;
      }
    }

    // Parallel cross-chunk suppression: OR kept rows' masks into the
    // register-resident suppressed vector (only words w > b matter).
    unsigned kw = keptw;
    while (kw) {
      int i = __ffs(kw) - 1;
      kw &= kw - 1;
      const unsigned* rp = bufc + i * 128;
      unsigned v0 = rp[lane];
      unsigned v1 = rp[lane + 32];
      unsigned v2 = rp[lane + 64];
      unsigned v3 = rp[lane + 96];
      if (lane > b)       rem0 |= v0;
      if (lane + 32 > b)  rem1 |= v1;
      if (lane + 64 > b)  rem2 |= v2;
      if (lane + 96 > b)  rem3 |= v3;
    }
    if (lane == 0) keptOut[b] = keptw;
  }
}

// -------------------------------------------------------------------------
// Kernel 4: out[p] = sortedRows[p] * kept[p]
// -------------------------------------------------------------------------
__global__ __launch_bounds__(256) void nms_write_kernel(
    const float* __restrict__ sortedRows, const unsigned* __restrict__ keptWords,
    float* __restrict__ out) {
  int p = blockIdx.x * 256 + threadIdx.x;
  if (p >= N_BOX) return;
  bool k = (keptWords[p >> 5] >> (p & 31)) & 1u;
  const float4* src = (const float4*)(sortedRows + (size_t)p * 8);
  float4 a = src[0], b = src[1];
  if (!k) {
    a = make_float4(0.f, 0.f, 0.f, 0.f);
    b = make_float4(0.f, 0.f, 0.f, 0.f);
  }
  float4* dst = (float4*)(out + (size_t)p * 8);
  dst[0] = a;
  dst[1] = b;
}

// -------------------------------------------------------------------------
// ws layout: [0,128KB) sorted rows | [128KB,128KB+2MB) mask | then 512B kept
// Every ws byte we read is rewritten every call -> graph-replay safe.
// -------------------------------------------------------------------------
extern "C" void kernel_launch(void* const* d_in, const int* in_sizes, int n_in,
                              void* d_out, int out_size, void* d_ws, size_t ws_size,
                              hipStream_t stream) {
  (void)in_sizes; (void)n_in; (void)out_size; (void)ws_size;
  const float* results = (const float*)d_in[0];
  char* ws = (char*)d_ws;
  float* sortedRows = (float*)ws;                                   // 131072 B
  unsigned short* mask16 = (unsigned short*)(ws + 131072);          // 2097152 B
  unsigned* keptWords = (unsigned*)(ws + 131072 + 2097152);         // 512 B
  float* out = (float*)d_out;

  nms_sort_kernel<<<1, 1024, 0, stream>>>(results, sortedRows);
  nms_mask_kernel<<<dim3(256, 256), 32, 0, stream>>>(sortedRows, mask16);
  nms_scan_kernel<<<1, 32, 0, stream>>>((const unsigned*)mask16, keptWords);
  nms_write_kernel<<<16, 256, 0, stream>>>(sortedRows, keptWords, out);
}